// GCN_47141561041507
// MI455X (gfx1250) — compile-verified
//
#include <hip/hip_runtime.h>

#define HH 128

typedef float v2f __attribute__((ext_vector_type(2)));
typedef float v8f __attribute__((ext_vector_type(8)));

// Guaranteed-native FP32 atomic add (no return -> STOREcnt, fire-and-forget).
// s_endpgm performs an implicit wait-idle, so completion is guaranteed at kernel end.
__device__ __forceinline__ void atomic_add_f32_native(float* addr, float val) {
    asm volatile("global_atomic_add_f32 %0, %1, off"
                 :
                 : "v"(addr), "v"(val)
                 : "memory");
}

// ---------------- degree ----------------
__global__ void deg_kernel(const int* __restrict__ dst, int* __restrict__ deg, int E) {
    int i = blockIdx.x * blockDim.x + threadIdx.x;
    if (i < E) atomicAdd(&deg[dst[i]], 1);
}

__global__ void invdeg_kernel(const int* __restrict__ deg, float* __restrict__ inv, int N) {
    int i = blockIdx.x * blockDim.x + threadIdx.x;
    if (i < N) {
        float d = (float)deg[i];
        inv[i] = 1.0f / fmaxf(d, 1.0f);
    }
}

// ---------------- edge scatter: agg[dst] += h[src] ----------------
// One wave32 per edge; lane L handles features [4L, 4L+4). Gather of h[src] is a
// fully-coalesced 512B/wave read (hits L2: h is 25.6MB << 192MB L2).
__global__ void scatter_kernel(const float* __restrict__ h, const int* __restrict__ src,
                               const int* __restrict__ dst, float* __restrict__ agg, int E) {
    int t = blockIdx.x * blockDim.x + threadIdx.x;
    int e = t >> 5;
    if (e >= E) return;
    int lane = t & 31;
    int s = src[e];
    int d = dst[e];
    const float4* hp = (const float4*)(h + (size_t)s * HH);
    float4 v = hp[lane];
    float* ap = agg + (size_t)d * HH + lane * 4;
    atomic_add_f32_native(ap + 0, v.x);
    atomic_add_f32_native(ap + 1, v.y);
    atomic_add_f32_native(ap + 2, v.z);
    atomic_add_f32_native(ap + 3, v.w);
}

// ---------------- WMMA GEMM: out = relu((agg * inv_deg) @ W + b) ----------------
// One wave per 16-row strip. 8 column tiles (H=128), K=128 via 32 x wmma 16x16x4 f32.
// A 16x4 fragment: rows = lane&15; lanes 0-15 hold K{0,1}, lanes 16-31 hold K{2,3}.
// B 4x16 fragment: cols = lane&15; same K split across lane halves.
// C/D: VGPR r holds (M=r, N=lane) for lanes 0-15 and (M=8+r, N=lane-16) for lanes 16-31.
__global__ void gemm_kernel(const float* __restrict__ A, const float* __restrict__ inv,
                            const float* __restrict__ W, const float* __restrict__ bias,
                            float* __restrict__ out, int nstrips, int N) {
    int lane  = threadIdx.x & 31;
    int wave  = threadIdx.x >> 5;
    int strip = blockIdx.x * (blockDim.x >> 5) + wave;
    if (strip >= nstrips) return;                 // wave-uniform: EXEC stays all-1s

    int col   = lane & 15;
    int khalf = (lane >> 4) << 1;                 // 0 or 2
    int mrow  = (strip << 4) + col;
    if (mrow >= N) mrow = N - 1;                  // clamp (no EXEC change)
    float id = inv[mrow];
    const float* arow = A + (size_t)mrow * HH;

    v8f acc[8];
#pragma unroll
    for (int n = 0; n < 8; ++n) acc[n] = (v8f){0.f,0.f,0.f,0.f,0.f,0.f,0.f,0.f};

    for (int kk = 0; kk < 32; ++kk) {
        int k0 = (kk << 2) + khalf;
        v2f a;
        a.x = arow[k0]     * id;                  // reference order: (agg*inv_deg) @ W
        a.y = arow[k0 + 1] * id;
        const float* w0 = W + (size_t)k0 * HH + col;
#pragma unroll
        for (int n = 0; n < 8; ++n) {
            v2f bf;
            bf.x = w0[n * 16];                    // W[k0,   n*16+col]
            bf.y = w0[n * 16 + HH];               // W[k0+1, n*16+col]
            acc[n] = __builtin_amdgcn_wmma_f32_16x16x4_f32(
                false, a, false, bf, (short)0, acc[n], false, false);
        }
    }

    int row0 = (strip << 4) + ((lane >> 4) << 3); // +8 for upper half-wave
#pragma unroll
    for (int n = 0; n < 8; ++n) {
        int c = (n << 4) + col;
        float bb = bias[c];
#pragma unroll
        for (int r = 0; r < 8; ++r) {
            int row = row0 + r;
            if (row < N) {
                float v = acc[n][r] + bb;
                out[(size_t)row * HH + c] = fmaxf(v, 0.0f);
            }
        }
    }
}

// ---------------- final layer: out = (agg * inv_deg) @ W3 + b3 (H=128 -> C=2) ----------------
__global__ void out_kernel(const float* __restrict__ A, const float* __restrict__ inv,
                           const float* __restrict__ W, const float* __restrict__ bias,
                           float* __restrict__ out, int N) {
    int i = blockIdx.x * blockDim.x + threadIdx.x;
    if (i >= N * 2) return;
    int v = i >> 1, c = i & 1;
    const float* a = A + (size_t)v * HH;
    float s = 0.f;
#pragma unroll 8
    for (int k = 0; k < HH; ++k) s = fmaf(a[k], W[k * 2 + c], s);
    out[i] = s * inv[v] + bias[c];
}

extern "C" void kernel_launch(void* const* d_in, const int* in_sizes, int n_in,
                              void* d_out, int out_size, void* d_ws, size_t ws_size,
                              hipStream_t stream) {
    const float* feat = (const float*)d_in[0];
    const int*   src  = (const int*)d_in[1];
    const int*   dst  = (const int*)d_in[2];
    const float* W0   = (const float*)d_in[3];
    const float* b0   = (const float*)d_in[4];
    const float* W1   = (const float*)d_in[5];
    const float* b1   = (const float*)d_in[6];
    const float* W2   = (const float*)d_in[7];
    const float* b2   = (const float*)d_in[8];
    const float* W3   = (const float*)d_in[9];
    const float* b3   = (const float*)d_in[10];

    const int N = in_sizes[0] / HH;
    const int E = in_sizes[1];

    char* ws = (char*)d_ws;
    size_t off = 0;
    int* deg = (int*)(ws + off);
    off += ((size_t)N * sizeof(int) + 255) & ~(size_t)255;
    float* inv = (float*)(ws + off);
    off += ((size_t)N * sizeof(float) + 255) & ~(size_t)255;
    float* bufA = (float*)(ws + off);               // h buffer
    off += (size_t)N * HH * sizeof(float);
    float* bufB = (float*)(ws + off);               // agg buffer

    const int nstrips = (N + 15) / 16;
    dim3 blk(256);
    dim3 grdE((E + 255) / 256);
    dim3 grdN((N + 255) / 256);
    dim3 grdScatter(((size_t)E * 32 + 255) / 256);
    dim3 grdGemm((nstrips + 7) / 8);                // 8 waves per block
    dim3 grdOut((N * 2 + 255) / 256);

    // degree / inverse degree
    hipMemsetAsync(deg, 0, (size_t)N * sizeof(int), stream);
    deg_kernel<<<grdE, blk, 0, stream>>>(dst, deg, E);
    invdeg_kernel<<<grdN, blk, 0, stream>>>(deg, inv, N);

    // layer 0: feat -> bufB (agg) -> bufA (h0)
    hipMemsetAsync(bufB, 0, (size_t)N * HH * sizeof(float), stream);
    scatter_kernel<<<grdScatter, blk, 0, stream>>>(feat, src, dst, bufB, E);
    gemm_kernel<<<grdGemm, blk, 0, stream>>>(bufB, inv, W0, b0, bufA, nstrips, N);

    // layer 1: bufA -> bufB -> bufA
    hipMemsetAsync(bufB, 0, (size_t)N * HH * sizeof(float), stream);
    scatter_kernel<<<grdScatter, blk, 0, stream>>>(bufA, src, dst, bufB, E);
    gemm_kernel<<<grdGemm, blk, 0, stream>>>(bufB, inv, W1, b1, bufA, nstrips, N);

    // layer 2: bufA -> bufB -> bufA
    hipMemsetAsync(bufB, 0, (size_t)N * HH * sizeof(float), stream);
    scatter_kernel<<<grdScatter, blk, 0, stream>>>(bufA, src, dst, bufB, E);
    gemm_kernel<<<grdGemm, blk, 0, stream>>>(bufB, inv, W2, b2, bufA, nstrips, N);

    // output layer: bufA -> bufB -> d_out (N x 2)
    hipMemsetAsync(bufB, 0, (size_t)N * HH * sizeof(float), stream);
    scatter_kernel<<<grdScatter, blk, 0, stream>>>(bufA, src, dst, bufB, E);
    out_kernel<<<grdOut, blk, 0, stream>>>(bufB, inv, W3, b3, (float*)d_out, N);
}